// CorrAttentionBias_47278999994623
// MI455X (gfx1250) — compile-verified
//
#include <hip/hip_runtime.h>
#include <hip/hip_bf16.h>
#include <stdint.h>

#define ALPHA 0.5f
#define BETA  0.1f
#define NEGV  -100000.0f

typedef float v4f __attribute__((ext_vector_type(4)));
typedef int   v4i __attribute__((ext_vector_type(4)));

constexpr int L_DIM = 2048;
constexpr int H_DIM = 16;
constexpr int TPB   = 256;          // 8 waves (wave32)
constexpr int VEC   = 4;            // float4 per thread
constexpr int TILE  = TPB * VEC;    // 1024 floats per column tile
constexpr int NTILE = L_DIM / TILE; // 2 tiles per row

__global__ __launch_bounds__(TPB)
void CorrAttentionBias_kernel(const float* __restrict__ attn,
                              const float* __restrict__ c_local,
                              const float* __restrict__ c_sink,
                              const int*   __restrict__ mask,
                              float* __restrict__ out)
{
    __shared__ float smem[2][TILE];

    const int tid = threadIdx.x;
    const int i   = blockIdx.x;        // row index in [0, L)
    const int bh  = blockIdx.y;        // b*H + h
    const int b   = bh >> 4;           // H == 16

    const size_t row_base = ((size_t)bh * L_DIM + (size_t)i) * L_DIM;
    const float* __restrict__ arow = attn + row_base;
    float* __restrict__ orow = out + row_base;

    // ---- per-row (block-uniform) scalars -> scalar loads ----
    const float s_i = c_sink[b * L_DIM + i];
    const bool  m_i = (mask[b * L_DIM + i] != 0);

    // edge term at j == i+1
    const float ep = (i + 1 < L_DIM) ? (ALPHA * c_local[b * L_DIM + i + 1]) : 0.0f;
    // edge term at j == i-1 (overwrite semantics of the reference)
    float em = 0.0f;
    if (i >= 1) {
        em = (i == 1 || i == L_DIM - 1) ? (ALPHA * c_local[b * L_DIM + i])
                                        : (ALPHA * c_local[b * L_DIM + i - 1]);
    }
    const int jp = i + 1;
    const int jm = i - 1;

    // ---- LDS byte offsets of this lane's staging slots ----
    // (flat-address low 32 bits == wave-relative LDS offset; HW adds LDS_BASE)
    const uint32_t lds0 = (uint32_t)(uintptr_t)&smem[0][tid * VEC];
    const uint32_t lds1 = (uint32_t)(uintptr_t)&smem[1][tid * VEC];

    // ---- prologue: async-copy tile 0 into LDS (ASYNCcnt = 1) ----
    {
        uint64_t g = (uint64_t)(uintptr_t)(arow + tid * VEC);
        asm volatile("global_load_async_to_lds_b128 %0, %1, off"
                     :: "v"(lds0), "v"(g) : "memory");
    }

    const float* __restrict__ csink_b = c_sink + b * L_DIM;
    const int*   __restrict__ mask_b  = mask   + b * L_DIM;

    for (int t = 0; t < NTILE; ++t) {
        if (t + 1 < NTILE) {
            // issue next tile's async copy, then wait for the older one only
            uint64_t g = (uint64_t)(uintptr_t)(arow + (t + 1) * TILE + tid * VEC);
            uint32_t l = (t & 1) ? lds0 : lds1;
            asm volatile("global_load_async_to_lds_b128 %0, %1, off"
                         :: "v"(l), "v"(g) : "memory");
            asm volatile("s_wait_asynccnt 0x1" ::: "memory");
        } else {
            asm volatile("s_wait_asynccnt 0x0" ::: "memory");
        }

        const int j0 = t * TILE + tid * VEC;

        // lane reads back exactly its own staged 16 B -> no barrier required
        v4f a  = *(const v4f*)&smem[t & 1][tid * VEC];
        v4f s  = *(const v4f*)&csink_b[j0];   // cache-resident, reused 32768x
        v4i mk = *(const v4i*)&mask_b[j0];

        v4f o;
        const float bsi = BETA * s_i;
#pragma unroll
        for (int k = 0; k < 4; ++k) {
            const int j = j0 + k;
            float v = fmaf(bsi, s[k], a[k]);
            if (j == jp) v += ep;
            if (j == jm) v += em;
            o[k] = (m_i | (mk[k] != 0)) ? NEGV : v;
        }

        // streaming output: b128 store with non-temporal hint (don't thrash L2)
        __builtin_nontemporal_store(o, (v4f*)&orow[j0]);
    }
}

extern "C" void kernel_launch(void* const* d_in, const int* in_sizes, int n_in,
                              void* d_out, int out_size, void* d_ws, size_t ws_size,
                              hipStream_t stream) {
    (void)n_in; (void)out_size; (void)d_ws; (void)ws_size;

    const float* attn    = (const float*)d_in[0];
    const float* c_local = (const float*)d_in[1];
    const float* c_sink  = (const float*)d_in[2];
    const int*   mask    = (const int*)d_in[3];
    float*       out     = (float*)d_out;

    const int B = in_sizes[1] / L_DIM;      // c_local is (B, L)
    dim3 grid(L_DIM, B * H_DIM);            // one block per (row, b*H+h)
    dim3 block(TPB);

    CorrAttentionBias_kernel<<<grid, block, 0, stream>>>(attn, c_local, c_sink, mask, out);
}